// STAN_91190745628885
// MI455X (gfx1250) — compile-verified
//
#include <hip/hip_runtime.h>
#include <math.h>

#define TSTEPS 4
#define BN     20000
#define EN     160000
#define NHEADS 4
#define GDIM   32

typedef __attribute__((ext_vector_type(2))) float v2f;
typedef __attribute__((ext_vector_type(8))) float v8f;

// ---------- helpers ----------
__device__ __forceinline__ unsigned f2ord(float f) {
  unsigned u = __float_as_uint(f);
  return (u & 0x80000000u) ? ~u : (u | 0x80000000u);
}
__device__ __forceinline__ float ord2f(unsigned u) {
  return (u & 0x80000000u) ? __uint_as_float(u & 0x7fffffffu) : __uint_as_float(~u);
}
__device__ __forceinline__ float sigmoidf(float x) { return 1.0f / (1.0f + expf(-x)); }
__device__ __forceinline__ float eluf(float x)     { return x > 0.0f ? x : (expf(x) - 1.0f); }

// ---------- WMMA projection: hp[h][n][o] = sum_k x[n][k] * W[h][o][k] ----------
// grid.x = BN/16 node tiles, 256 threads = 8 waves; wave -> (head, o-tile of 16)
__global__ void proj_wmma_kernel(const float* __restrict__ x, const float* __restrict__ W,
                                 float* __restrict__ hp, int K) {
  __shared__ float xs[16 * 128];   // max K = 128
  const int n0 = blockIdx.x * 16;
  for (int i = threadIdx.x; i < 16 * K; i += blockDim.x)
    xs[i] = x[(size_t)n0 * K + i];          // rows contiguous in both
  __syncthreads();

  const int wave = threadIdx.x >> 5;
  const int lane = threadIdx.x & 31;
  const int h    = wave >> 1;
  const int o0   = (wave & 1) * 16;
  const int m    = lane & 15;
  const int hi   = lane >> 4;               // lane half selects K=0/1 vs K=2/3

  const float* Wh = W + (size_t)h * 32 * K; // W[h][o][k]
  v8f c = {};
  for (int k0 = 0; k0 < K; k0 += 4) {
    const int ka = k0 + hi * 2;
    v2f a, b;
    a[0] = xs[m * K + ka];
    a[1] = xs[m * K + ka + 1];
    const float* wp = Wh + (size_t)(o0 + m) * K + ka;
    b[0] = wp[0];
    b[1] = wp[1];
    c = __builtin_amdgcn_wmma_f32_16x16x4_f32(false, a, false, b, (short)0, c, false, false);
  }
  // D[mm][nn]: mm = v + hi*8, nn = m
  float* op = hp + ((size_t)h * BN + n0) * 32 + o0 + m;
#pragma unroll
  for (int v = 0; v < 8; ++v) {
    const int mm = v + hi * 8;
    op[(size_t)mm * 32] = c[v];
  }
}

// ---------- WMMA GRU hidden matmul: gh[n][j] = sum_g h[n][g] * W_hh[j][g] ----------
// grid.x = BN/16, 192 threads = 6 waves; wave -> j-tile of 16 (N = 96)
__global__ void gh_wmma_kernel(const float* __restrict__ h, const float* __restrict__ Whh,
                               float* __restrict__ gh) {
  __shared__ float xs[16 * 32];
  const int n0 = blockIdx.x * 16;
  for (int i = threadIdx.x; i < 16 * 32; i += blockDim.x)
    xs[i] = h[(size_t)n0 * 32 + i];
  __syncthreads();

  const int wave = threadIdx.x >> 5;
  const int lane = threadIdx.x & 31;
  const int j0   = wave * 16;
  const int m    = lane & 15;
  const int hi   = lane >> 4;

  v8f c = {};
  for (int k0 = 0; k0 < 32; k0 += 4) {
    const int ka = k0 + hi * 2;
    v2f a, b;
    a[0] = xs[m * 32 + ka];
    a[1] = xs[m * 32 + ka + 1];
    const float* wp = Whh + (size_t)(j0 + m) * 32 + ka;
    b[0] = wp[0];
    b[1] = wp[1];
    c = __builtin_amdgcn_wmma_f32_16x16x4_f32(false, a, false, b, (short)0, c, false, false);
  }
  float* op = gh + (size_t)n0 * 96 + j0 + m;
#pragma unroll
  for (int v = 0; v < 8; ++v) {
    const int mm = v + hi * 8;
    op[(size_t)mm * 96] = c[v];
  }
}

// ---------- per-(head,node) attention scores ----------
__global__ void scores_kernel(const float* __restrict__ hp, const float* __restrict__ a,
                              float* __restrict__ es, float* __restrict__ ed) {
  const int idx = blockIdx.x * blockDim.x + threadIdx.x; // h*BN + n
  if (idx >= NHEADS * BN) return;
  const int h = idx / BN;
  const float* row = hp + (size_t)idx * 32;
  const float* ah  = a + h * 64;
  float s = 0.f, d = 0.f;
#pragma unroll
  for (int o = 0; o < 32; ++o) {
    const float v = row[o];
    s += v * ah[o];
    d += v * ah[32 + o];
  }
  es[idx] = s;
  ed[idx] = d;
}

// ---------- scalar init ----------
__global__ void init_scalars_kernel(float* scal) {
  if (threadIdx.x < NHEADS) {
    ((unsigned*)scal)[threadIdx.x] = 0u;   // identity for ordered-uint max
    scal[NHEADS + threadIdx.x]     = 0.f;  // Z sums
  }
}

// ---------- edge pass 1: e = leaky_relu(es[src]+ed[dst]); global per-head max ----------
__global__ void edge_max_kernel(const float* __restrict__ es, const float* __restrict__ ed,
                                const int* __restrict__ src, const int* __restrict__ dst,
                                float* __restrict__ ew, unsigned* __restrict__ maxu) {
  __shared__ float red[256];
  const int gid = blockIdx.x * 256 + threadIdx.x; // h*EN + e  (EN % 256 == 0)
  const int h = gid / EN;
  const int e = gid - h * EN;
  float ev = es[h * BN + src[e]] + ed[h * BN + dst[e]];
  ev = ev > 0.f ? ev : 0.01f * ev;  // leaky_relu(0.01)
  ew[gid] = ev;
  red[threadIdx.x] = ev;
  __syncthreads();
  for (int s = 128; s > 0; s >>= 1) {
    if (threadIdx.x < s) red[threadIdx.x] = fmaxf(red[threadIdx.x], red[threadIdx.x + s]);
    __syncthreads();
  }
  if (threadIdx.x == 0) atomicMax(&maxu[h], f2ord(red[0]));
}

// ---------- edge pass 2: w = exp(e - max); Z += w; scatter w*h_src into acc[dst] ----------
__global__ void edge_scatter_kernel(const float* __restrict__ ew, const unsigned* __restrict__ maxu,
                                    float* __restrict__ Z, const float* __restrict__ hp,
                                    const int* __restrict__ src, const int* __restrict__ dst,
                                    float* __restrict__ acc) {
  __shared__ float red[256];
  const int gid = blockIdx.x * 256 + threadIdx.x;
  const int h = gid / EN;
  const int e = gid - h * EN;
  const float w = expf(ew[gid] - ord2f(maxu[h]));
  red[threadIdx.x] = w;
  __syncthreads();
  for (int s = 128; s > 0; s >>= 1) {
    if (threadIdx.x < s) red[threadIdx.x] += red[threadIdx.x + s];
    __syncthreads();
  }
  if (threadIdx.x == 0) atomicAdd(&Z[h], red[0]);

  const float4* hs = (const float4*)(hp + ((size_t)h * BN + src[e]) * 32);
  float* op = acc + ((size_t)h * BN + dst[e]) * 32;
#pragma unroll
  for (int q = 0; q < 8; ++q) {
    const float4 v = hs[q];
    atomicAdd(op + q * 4 + 0, w * v.x);
    atomicAdd(op + q * 4 + 1, w * v.y);
    atomicAdd(op + q * 4 + 2, w * v.z);
    atomicAdd(op + q * 4 + 3, w * v.w);
  }
}

// ---------- normalize + ELU + head-concat transpose: xc[n][h*32+o] ----------
__global__ void norm_elu_kernel(const float* __restrict__ acc, const float* __restrict__ Z,
                                float* __restrict__ xc) {
  const int idx = blockIdx.x * blockDim.x + threadIdx.x; // n*128 + j
  if (idx >= BN * 128) return;
  const int n = idx >> 7;
  const int j = idx & 127;
  const int h = j >> 5;
  const int o = j & 31;
  const float v = acc[((size_t)h * BN + n) * 32 + o] / Z[h];
  xc[idx] = eluf(v);
}

// ---------- node mean over 128 features ----------
__global__ void mean_kernel(const float* __restrict__ xc, float* __restrict__ cur) {
  const int n = blockIdx.x * blockDim.x + threadIdx.x;
  if (n >= BN) return;
  const float* r = xc + (size_t)n * 128;
  float s = 0.f;
#pragma unroll 4
  for (int i = 0; i < 128; ++i) s += r[i];
  cur[n] = s * (1.0f / 128.0f);
}

// ---------- GRU cell (in-place hidden update) ----------
__global__ void gru_kernel(const float* __restrict__ cur, const float* __restrict__ gh,
                           const float* __restrict__ W_ih, const float* __restrict__ b_ih,
                           const float* __restrict__ b_hh, float* __restrict__ hstate) {
  const int idx = blockIdx.x * blockDim.x + threadIdx.x; // n*32 + g
  if (idx >= BN * GDIM) return;
  const int n = idx >> 5;
  const int g = idx & 31;
  const float cv = cur[n];
  const float ir  = cv * W_ih[g]      + b_ih[g];
  const float iz  = cv * W_ih[32 + g] + b_ih[32 + g];
  const float in_ = cv * W_ih[64 + g] + b_ih[64 + g];
  const float* gr = gh + (size_t)n * 96;
  const float hr = gr[g]      + b_hh[g];
  const float hz = gr[32 + g] + b_hh[32 + g];
  const float hn = gr[64 + g] + b_hh[64 + g];
  const float r  = sigmoidf(ir + hr);
  const float z  = sigmoidf(iz + hz);
  const float nn = tanhf(in_ + r * hn);
  hstate[idx] = (1.0f - z) * nn + z * hstate[idx];
}

// ---------- prediction heads + SIR physics (last step only) ----------
__global__ void heads_kernel(const float* __restrict__ hstate,
                             const float* __restrict__ cI_t, const float* __restrict__ cR_t,
                             const float* __restrict__ Nv, const float* __restrict__ Iv,
                             const float* __restrict__ Rv,
                             const float* __restrict__ W_I, const float* __restrict__ b_I,
                             const float* __restrict__ W_R, const float* __restrict__ b_R,
                             const float* __restrict__ W_sir, const float* __restrict__ b_sir,
                             float* __restrict__ out) {
  const int n = blockIdx.x * blockDim.x + threadIdx.x;
  if (n >= BN) return;
  float hc[34];
  const float* hr = hstate + (size_t)n * 32;
#pragma unroll
  for (int g = 0; g < 32; ++g) hc[g] = hr[g];
  hc[32] = cI_t[n];
  hc[33] = cR_t[n];

  float* predI = out;
  float* predR = out + (size_t)BN * 5;
  float* phyI  = out + (size_t)BN * 10;
  float* phyR  = out + (size_t)BN * 11;
  float* hfin  = out + (size_t)BN * 12;

#pragma unroll
  for (int p = 0; p < 5; ++p) {
    float sI = b_I[p], sR = b_R[p];
    const float* wi = W_I + p * 34;
    const float* wr = W_R + p * 34;
#pragma unroll
    for (int j = 0; j < 34; ++j) { sI += hc[j] * wi[j]; sR += hc[j] * wr[j]; }
    predI[(size_t)n * 5 + p] = sI;
    predR[(size_t)n * 5 + p] = sR;
  }
  float s0 = b_sir[0], s1 = b_sir[1];
#pragma unroll
  for (int j = 0; j < 34; ++j) { s0 += hc[j] * W_sir[j]; s1 += hc[j] * W_sir[34 + j]; }
  const float al = sigmoidf(s0), be = sigmoidf(s1);
  const float Nf = Nv[n], If = Iv[n], Rf = Rv[n];
  const float S  = fmaxf(Nf - If - Rf, 0.0f);
  phyI[n] = al * If * (S / Nf) - be * If;
  phyR[n] = be * If;
#pragma unroll
  for (int g = 0; g < 32; ++g) hfin[(size_t)n * 32 + g] = hr[g];
}

// ---------- workspace layout (floats) ----------
#define HP_OFF   0u            // NHEADS*BN*32 = 2,560,000
#define ACC_OFF  2560000u      // 2,560,000
#define XC_OFF   5120000u      // BN*128 = 2,560,000
#define ES_OFF   7680000u      // 80,000
#define ED_OFF   7760000u      // 80,000
#define EW_OFF   7840000u      // NHEADS*EN = 640,000
#define HS_OFF   8480000u      // BN*32 = 640,000
#define GH_OFF   9120000u      // BN*96 = 1,920,000
#define CUR_OFF  11040000u     // 20,000
#define SC_OFF   11060000u     // 8 (4 maxu as uint + 4 Z)

extern "C" void kernel_launch(void* const* d_in, const int* in_sizes, int n_in,
                              void* d_out, int out_size, void* d_ws, size_t ws_size,
                              hipStream_t stream) {
  const float* dynamic = (const float*)d_in[0];
  const float* cI      = (const float*)d_in[1];
  const float* cR      = (const float*)d_in[2];
  const float* Nv      = (const float*)d_in[3];
  const float* Iv      = (const float*)d_in[4];
  const float* Rv      = (const float*)d_in[5];
  const int*   ei      = (const int*)d_in[6];
  const float* W1      = (const float*)d_in[7];
  const float* a1      = (const float*)d_in[8];
  const float* W2      = (const float*)d_in[9];
  const float* a2      = (const float*)d_in[10];
  const float* W_ih    = (const float*)d_in[11];
  const float* W_hh    = (const float*)d_in[12];
  const float* b_ih    = (const float*)d_in[13];
  const float* b_hh    = (const float*)d_in[14];
  const float* W_I     = (const float*)d_in[15];
  const float* b_I     = (const float*)d_in[16];
  const float* W_R     = (const float*)d_in[17];
  const float* b_R     = (const float*)d_in[18];
  const float* W_sir   = (const float*)d_in[19];
  const float* b_sir   = (const float*)d_in[20];
  float* out = (float*)d_out;

  float* ws   = (float*)d_ws;
  float* hp   = ws + HP_OFF;
  float* acc  = ws + ACC_OFF;
  float* xc   = ws + XC_OFF;
  float* es   = ws + ES_OFF;
  float* ed   = ws + ED_OFF;
  float* ew   = ws + EW_OFF;
  float* hs   = ws + HS_OFF;
  float* gh   = ws + GH_OFF;
  float* cur  = ws + CUR_OFF;
  float* scal = ws + SC_OFF;

  const int* srcI = ei;
  const int* dstI = ei + EN;

  const int ntile   = BN / 16;                       // 1250
  const int nEblk   = (NHEADS * EN) / 256;           // 2500
  const int nScore  = (NHEADS * BN + 255) / 256;     // 313
  const int nElem   = (BN * 128) / 256;              // 10000
  const int nGru    = (BN * GDIM + 255) / 256;       // 2500
  const int nNode   = (BN + 255) / 256;              // 79

  hipMemsetAsync(hs, 0, (size_t)BN * GDIM * sizeof(float), stream);

  for (int t = 0; t < TSTEPS; ++t) {
    const float* layer_x[2] = { dynamic + (size_t)t * BN * 32, xc };
    const int    layer_K[2] = { 32, 128 };
    const float* layer_W[2] = { W1, W2 };
    const float* layer_a[2] = { a1, a2 };

    for (int l = 0; l < 2; ++l) {
      hipMemsetAsync(acc, 0, (size_t)NHEADS * BN * 32 * sizeof(float), stream);
      init_scalars_kernel<<<1, 32, 0, stream>>>(scal);
      proj_wmma_kernel<<<ntile, 256, 0, stream>>>(layer_x[l], layer_W[l], hp, layer_K[l]);
      scores_kernel<<<nScore, 256, 0, stream>>>(hp, layer_a[l], es, ed);
      edge_max_kernel<<<nEblk, 256, 0, stream>>>(es, ed, srcI, dstI, ew, (unsigned*)scal);
      edge_scatter_kernel<<<nEblk, 256, 0, stream>>>(ew, (const unsigned*)scal, scal + NHEADS,
                                                     hp, srcI, dstI, acc);
      norm_elu_kernel<<<nElem, 256, 0, stream>>>(acc, scal + NHEADS, xc);
    }

    mean_kernel<<<nNode, 256, 0, stream>>>(xc, cur);
    gh_wmma_kernel<<<ntile, 192, 0, stream>>>(hs, W_hh, gh);
    gru_kernel<<<nGru, 256, 0, stream>>>(cur, gh, W_ih, b_ih, b_hh, hs);
  }

  heads_kernel<<<nNode, 256, 0, stream>>>(hs, cI + (size_t)(TSTEPS - 1) * BN,
                                          cR + (size_t)(TSTEPS - 1) * BN,
                                          Nv, Iv, Rv, W_I, b_I, W_R, b_R, W_sir, b_sir, out);
}